// SparseConvTranspose_40819369181594
// MI455X (gfx1250) — compile-verified
//
#include <hip/hip_runtime.h>

typedef __attribute__((ext_vector_type(2))) float v2f;
typedef __attribute__((ext_vector_type(8))) float v8f;

#define CIN   16
#define COUT  32
#define TAPS  27
#define WAVES_PER_BLOCK 8
#define TILES_PER_WAVE  2
#define BLOCK 256
// LDS weight layout: [t][q][half][n][kk] -> t*512 + q*128 + half*64 + n*4 + kk
#define WLDS (TAPS * 512)

__global__ __launch_bounds__(BLOCK) void sconvT_wmma_f32(
    const float* __restrict__ inp,   // [N,16]
    const float* __restrict__ wgt,   // [27,16,32] = (t, cin, cout)
    const float* __restrict__ bias,  // [32]
    const int*   __restrict__ nbr,   // [N,27]
    float* __restrict__ out)         // [N,32]
{
    __shared__ float lw[WLDS];       // 55296 bytes

    const int tid = threadIdx.x;

    // Stage weights: coalesced global reads, permuted LDS writes so each
    // lane's B fragment is one contiguous b64 in LDS.
    for (int i = tid; i < WLDS; i += BLOCK) {
        const int t    = i >> 9;
        const int cin  = (i >> 5) & 15;
        const int cout = i & 31;
        const int q    = cin >> 2;
        const int kk   = cin & 3;
        const int half = cout >> 4;
        const int nn   = cout & 15;
        lw[t * 512 + q * 128 + half * 64 + nn * 4 + kk] = wgt[i];
    }
    __syncthreads();

    const int lane  = tid & 31;
    const int m     = lane & 15;     // point-row within tile (A matrix M)
    const int hi    = lane >> 4;     // selects K pair (A/B) and M+8 rows (C/D)
    const int wave  = blockIdx.x * WAVES_PER_BLOCK + (tid >> 5);
    const int tileA = wave * TILES_PER_WAVE;
    const int tileB = tileA + 1;

    const int* __restrict__ nrowA = nbr + (long)(tileA * 16 + m) * TAPS;
    const int* __restrict__ nrowB = nbr + (long)(tileB * 16 + m) * TAPS;

    v8f accA0 = {}, accA1 = {};      // tile A: cout 0..15 / 16..31
    v8f accB0 = {}, accB1 = {};      // tile B

    for (int t = 0; t < TAPS; ++t) {
        const int ia = nrowA[t];
        const int ib = nrowB[t];
        const unsigned ba = __builtin_amdgcn_ballot_w32(ia >= 0);
        const unsigned bb = __builtin_amdgcn_ballot_w32(ib >= 0);
        if ((ba | bb) == 0u) continue;          // uniform: tap dead for both tiles

        // A fragments: chunk q -> cin {4q + 2*hi, 4q + 2*hi + 1}
        v2f aA[4], aB[4];
        aA[0] = (v2f){0.f, 0.f}; aA[1] = aA[0]; aA[2] = aA[0]; aA[3] = aA[0];
        aB[0] = aA[0];           aB[1] = aA[0]; aB[2] = aA[0]; aB[3] = aA[0];
        if (ia >= 0) {
            const float* p = inp + (long)ia * CIN + (hi << 1);
            aA[0] = *(const v2f*)(p);
            aA[1] = *(const v2f*)(p + 4);
            aA[2] = *(const v2f*)(p + 8);
            aA[3] = *(const v2f*)(p + 12);
        }
        if (ib >= 0) {
            const float* p = inp + (long)ib * CIN + (hi << 1);
            aB[0] = *(const v2f*)(p);
            aB[1] = *(const v2f*)(p + 4);
            aB[2] = *(const v2f*)(p + 8);
            aB[3] = *(const v2f*)(p + 12);
        }

        // B fragments once per tap, shared by both tiles.
        const float* wl = lw + t * 512 + (m << 2) + (hi << 1);
        v2f b0[4], b1[4];
        #pragma unroll
        for (int q = 0; q < 4; ++q) {
            b0[q] = *(const v2f*)(wl + q * 128);        // cout half 0
            b1[q] = *(const v2f*)(wl + q * 128 + 64);   // cout half 1
        }

        // Uniform branches (EXEC stays all-1s for WMMA).
        if (ba) {
            #pragma unroll
            for (int q = 0; q < 4; ++q) {
                accA0 = __builtin_amdgcn_wmma_f32_16x16x4_f32(
                            false, aA[q], false, b0[q], (short)0, accA0, false, false);
                accA1 = __builtin_amdgcn_wmma_f32_16x16x4_f32(
                            false, aA[q], false, b1[q], (short)0, accA1, false, false);
            }
        }
        if (bb) {
            #pragma unroll
            for (int q = 0; q < 4; ++q) {
                accB0 = __builtin_amdgcn_wmma_f32_16x16x4_f32(
                            false, aB[q], false, b0[q], (short)0, accB0, false, false);
                accB1 = __builtin_amdgcn_wmma_f32_16x16x4_f32(
                            false, aB[q], false, b1[q], (short)0, accB1, false, false);
            }
        }
    }

    // D layout: VGPR r, lane L -> point tile*16 + r + hi*8, cout half*16 + (L&15)
    const float bc0 = bias[m];
    const float bc1 = bias[16 + m];

    float* orowA = out + (long)(tileA * 16 + hi * 8) * COUT + m;
    float* orowB = out + (long)(tileB * 16 + hi * 8) * COUT + m;
    #pragma unroll
    for (int r = 0; r < 8; ++r) {
        orowA[r * COUT]      = accA0[r] + bc0;
        orowA[r * COUT + 16] = accA1[r] + bc1;
        orowB[r * COUT]      = accB0[r] + bc0;
        orowB[r * COUT + 16] = accB1[r] + bc1;
    }
}

extern "C" void kernel_launch(void* const* d_in, const int* in_sizes, int n_in,
                              void* d_out, int out_size, void* d_ws, size_t ws_size,
                              hipStream_t stream) {
    const float* inp  = (const float*)d_in[0];   // [N,16]
    const float* wgt  = (const float*)d_in[1];   // [3,3,3,16,32]
    const float* bias = (const float*)d_in[2];   // [32]
    const int*   nbr  = (const int*)d_in[3];     // [N,27]
    float* out = (float*)d_out;

    const int n      = in_sizes[0] / CIN;                         // 262144
    const int tiles  = n / 16;                                    // 16384
    const int blocks = tiles / (WAVES_PER_BLOCK * TILES_PER_WAVE);// 1024

    hipLaunchKernelGGL(sconvT_wmma_f32, dim3(blocks), dim3(BLOCK), 0, stream,
                       inp, wgt, bias, nbr, out);
}